// Scorer_21268678050183
// MI455X (gfx1250) — compile-verified
//
#include <hip/hip_runtime.h>

typedef __attribute__((ext_vector_type(2))) float v2f;
typedef __attribute__((ext_vector_type(8))) float v8f;

#define EPSGN 1e-5f

__device__ __forceinline__ v8f wmma4(v2f a, v2f b, v8f c) {
  // D = A(16x4,f32) * B(4x16,f32) + C(16x16,f32)
  return __builtin_amdgcn_wmma_f32_16x16x4_f32(false, a, false, b, (short)0, c,
                                               false, false);
}

// monotone mapping f32 -> u32 (ascending)
__device__ __forceinline__ unsigned monof(float v) {
  unsigned u = __float_as_uint(v);
  return (u & 0x80000000u) ? ~u : (u | 0x80000000u);
}

// ascending bitonic sort of 256 u64 keys in LDS, 256 threads
__device__ __forceinline__ void bitonic256(unsigned long long* keys, int tid) {
  for (int k = 2; k <= 256; k <<= 1) {
    for (int j = k >> 1; j > 0; j >>= 1) {
      __syncthreads();
      int ixj = tid ^ j;
      if (ixj > tid) {
        unsigned long long a = keys[tid], b = keys[ixj];
        bool asc = (tid & k) == 0;
        bool sw = asc ? (a > b) : (a < b);
        if (sw) { keys[tid] = b; keys[ixj] = a; }
      }
    }
  }
  __syncthreads();
}

// ---------------- Kernel A: batch-independent precompute ----------------
__global__ __launch_bounds__(256) void k_precompute(
    const float* __restrict__ points, const float* __restrict__ ref_feats,
    const float* __restrict__ src_feats, float* __restrict__ w_hdr,
    float* __restrict__ w_x2, float* __restrict__ w_refn,
    float* __restrict__ w_sim, float* __restrict__ w_sms,
    int* __restrict__ w_r2si, int* __restrict__ w_s2ri,
    int* __restrict__ w_permf) {
  __shared__ float rp[128][3];
  __shared__ float n2[128];
  __shared__ float cms[4];  // centroid xyz, max-norm scale
  __shared__ float sim[128][129];
  __shared__ float vmax[256];
  __shared__ unsigned long long keys[256];

  int tid = threadIdx.x;
  if (tid < 128) {
    float px = points[tid * 3 + 0], py = points[tid * 3 + 1],
          pz = points[tid * 3 + 2];
    rp[tid][0] = px; rp[tid][1] = py; rp[tid][2] = pz;
    n2[tid] = px * px + py * py + pz * pz;
  }
  __syncthreads();
  if (tid < 4) {
    if (tid < 3) {
      float s = 0.f;
      for (int n = 0; n < 128; ++n) s += rp[n][tid];
      cms[tid] = s * (1.f / 128.f);
    } else {
      float mx = 0.f;
      for (int n = 0; n < 128; ++n) mx = fmaxf(mx, sqrtf(n2[n]));
      cms[3] = mx;
    }
  }
  __syncthreads();
  if (tid < 128) {
    float inv = 1.f / cms[3];
    float a0 = (rp[tid][0] - cms[0]) * inv;
    float a1 = (rp[tid][1] - cms[1]) * inv;
    float a2 = (rp[tid][2] - cms[2]) * inv;
    w_refn[tid * 4 + 0] = a0; w_refn[tid * 4 + 1] = a1;
    w_refn[tid * 4 + 2] = a2; w_refn[tid * 4 + 3] = 0.f;
    w_x2[tid] = a0 * a0 + a1 * a1 + a2 * a2;
  }
  if (tid < 4) w_hdr[tid] = cms[tid];

  // sim = ref_feats(128x32) @ src_feats(128x32)^T via WMMA f32 16x16x4
  int lane = tid & 31, wave = tid >> 5;
  int mn = lane & 15;           // M for A-frag, N for B-frag
  int kb = (lane >> 4) << 1;    // K pair base per half-wave
  int half = lane >> 4;
  for (int t = 0; t < 8; ++t) {
    int tile = wave * 8 + t;
    int tm = tile >> 3, tn = tile & 7;
    v8f acc = {0.f, 0.f, 0.f, 0.f, 0.f, 0.f, 0.f, 0.f};
#pragma unroll
    for (int k0 = 0; k0 < 32; k0 += 4) {
      v2f a, b;
      a.x = ref_feats[(tm * 16 + mn) * 32 + k0 + kb];
      a.y = ref_feats[(tm * 16 + mn) * 32 + k0 + kb + 1];
      b.x = src_feats[(tn * 16 + mn) * 32 + k0 + kb];
      b.y = src_feats[(tn * 16 + mn) * 32 + k0 + kb + 1];
      acc = wmma4(a, b, acc);
    }
#pragma unroll
    for (int r = 0; r < 8; ++r) {
      int mm = tm * 16 + r + 8 * half;
      int nn = tn * 16 + mn;
      sim[mm][nn] = acc[r];
      w_sim[mm * 128 + nn] = acc[r];
    }
  }
  __syncthreads();
  // row / col argmax of sim (first-max tie rule, matching jnp.argmax)
  if (tid < 128) {
    float best = sim[tid][0]; int bi = 0;
    for (int j = 1; j < 128; ++j) {
      float v = sim[tid][j];
      if (v > best) { best = v; bi = j; }
    }
    vmax[tid] = best; w_r2si[tid] = bi;
  } else {
    int j = tid - 128;
    float best = sim[0][j]; int bi = 0;
    for (int i = 1; i < 128; ++i) {
      float v = sim[i][j];
      if (v > best) { best = v; bi = i; }
    }
    vmax[tid] = best; w_s2ri[j] = bi;
  }
  __syncthreads();
  // stable descending argsort of max_sim (batch-independent perm_f)
  keys[tid] = ((unsigned long long)(~monof(vmax[tid])) << 32) | (unsigned)tid;
  bitonic256(keys, tid);
  int idx = (int)(keys[tid] & 0xffffffffull);
  w_permf[tid] = idx;
  w_sms[tid] = vmax[idx];
}

// ---------------- Kernel B: per-batch scores -> joint(512) ----------------
__global__ __launch_bounds__(256) void k_batch(
    const float* __restrict__ points, const float* __restrict__ trans,
    const float* __restrict__ w_hdr, const float* __restrict__ w_x2,
    const float* __restrict__ w_refn, const float* __restrict__ w_sim,
    const float* __restrict__ w_sms, const int* __restrict__ w_r2si,
    const int* __restrict__ w_s2ri, const int* __restrict__ w_permf,
    float* __restrict__ w_joint) {
  __shared__ float refn[128][4];
  __shared__ float srcn[128][4];
  __shared__ float x2s[128], y2s[128];
  __shared__ float dist[128][129];
  __shared__ float rmax[128]; __shared__ int rarg[128];
  __shared__ float cmax[128]; __shared__ int carg[128];
  __shared__ float nd[256], ns[256];
  __shared__ unsigned long long keys[256];
  __shared__ float cms[4];

  int b = blockIdx.x;
  int tid = threadIdx.x;
  if (tid < 4) cms[tid] = w_hdr[tid];
  if (tid < 128) {
    refn[tid][0] = w_refn[tid * 4 + 0];
    refn[tid][1] = w_refn[tid * 4 + 1];
    refn[tid][2] = w_refn[tid * 4 + 2];
    refn[tid][3] = 0.f;
    x2s[tid] = w_x2[tid];
  }
  __syncthreads();
  if (tid < 128) {
    const float* T = trans + b * 16;
    float px = points[(128 + tid) * 3 + 0];
    float py = points[(128 + tid) * 3 + 1];
    float pz = points[(128 + tid) * 3 + 2];
    float sx = T[0] * px + T[1] * py + T[2] * pz + T[3];
    float sy = T[4] * px + T[5] * py + T[6] * pz + T[7];
    float sz = T[8] * px + T[9] * py + T[10] * pz + T[11];
    float inv = 1.f / cms[3];
    sx = (sx - cms[0]) * inv;
    sy = (sy - cms[1]) * inv;
    sz = (sz - cms[2]) * inv;
    srcn[tid][0] = sx; srcn[tid][1] = sy; srcn[tid][2] = sz; srcn[tid][3] = 0.f;
    y2s[tid] = sx * sx + sy * sy + sz * sz;
  }
  __syncthreads();

  // xy = ref_n(128x4,pad) @ src_n^T(4x128) via WMMA; fuse exp into writeback
  int lane = tid & 31, wave = tid >> 5;
  int mn = lane & 15;
  int kb = (lane >> 4) << 1;
  int half = lane >> 4;
  for (int t = 0; t < 8; ++t) {
    int tile = wave * 8 + t;
    int tm = tile >> 3, tn = tile & 7;
    v2f a, bb;
    a.x = refn[tm * 16 + mn][kb];
    a.y = refn[tm * 16 + mn][kb + 1];
    bb.x = srcn[tn * 16 + mn][kb];
    bb.y = srcn[tn * 16 + mn][kb + 1];
    v8f acc = {0.f, 0.f, 0.f, 0.f, 0.f, 0.f, 0.f, 0.f};
    acc = wmma4(a, bb, acc);
#pragma unroll
    for (int r = 0; r < 8; ++r) {
      int mm = tm * 16 + r + 8 * half;
      int nn = tn * 16 + mn;
      dist[mm][nn] = __expf(-(x2s[mm] + y2s[nn] - 2.f * acc[r]));
    }
  }
  __syncthreads();

  // row / col argmax of dist
  if (tid < 128) {
    float best = dist[tid][0]; int bi = 0;
    for (int j = 1; j < 128; ++j) {
      float v = dist[tid][j];
      if (v > best) { best = v; bi = j; }
    }
    rmax[tid] = best; rarg[tid] = bi;
  } else {
    int j = tid - 128;
    float best = dist[0][j]; int bi = 0;
    for (int i = 1; i < 128; ++i) {
      float v = dist[i][j];
      if (v > best) { best = v; bi = i; }
    }
    cmax[j] = best; carg[j] = bi;
  }
  __syncthreads();

  if (tid < 128) {
    nd[tid] = rmax[tid];
    ns[tid] = w_sim[tid * 128 + rarg[tid]];
  } else {
    int j = tid - 128;
    nd[tid] = cmax[j];
    ns[tid] = w_sim[carg[j] * 128 + j];
  }
  __syncthreads();

  // stable descending argsort of nearest_dist
  keys[tid] = ((unsigned long long)(~monof(nd[tid])) << 32) | (unsigned)tid;
  bitonic256(keys, tid);
  int idx = (int)(keys[tid] & 0xffffffffull);
  float nearest_score = nd[idx] * ns[idx];

  // maxsimi_score using batch-independent perm_f / sorted max_sim
  int pj = w_permf[tid];
  float md = (pj < 128) ? dist[pj][w_r2si[pj]]
                        : dist[w_s2ri[pj - 128]][pj - 128];
  float maxsimi_score = w_sms[tid] * md;

  w_joint[b * 512 + tid] = nearest_score;
  w_joint[b * 512 + 256 + tid] = maxsimi_score;
}

// ---------------- Kernel C: MLP, 16 batch rows per block ----------------
__global__ __launch_bounds__(256) void k_mlp(
    const float* __restrict__ w_joint, const float* __restrict__ W1,
    const float* __restrict__ b1, const float* __restrict__ g1,
    const float* __restrict__ beta1, const float* __restrict__ W2,
    const float* __restrict__ b2, const float* __restrict__ g2,
    const float* __restrict__ beta2, const float* __restrict__ W3,
    const float* __restrict__ b3, float* __restrict__ out) {
  __shared__ float jt[16][512];
  __shared__ float h1[16][256];
  __shared__ float h2[16][128];

  int blk = blockIdx.x;
  int tid = threadIdx.x;
  for (int i = tid; i < 16 * 512; i += 256)
    jt[i >> 9][i & 511] = w_joint[blk * 16 * 512 + i];
  __syncthreads();

  int lane = tid & 31, wave = tid >> 5;
  int mn = lane & 15;
  int kb = (lane >> 4) << 1;
  int half = lane >> 4;

  // GEMM1: (16x512) @ W1(512x256), 16 N-tiles, 2 per wave
  for (int s = 0; s < 2; ++s) {
    int tn = wave * 2 + s;
    v8f acc = {0.f, 0.f, 0.f, 0.f, 0.f, 0.f, 0.f, 0.f};
    for (int k0 = 0; k0 < 512; k0 += 4) {
      v2f a, bb;
      a.x = jt[mn][k0 + kb];
      a.y = jt[mn][k0 + kb + 1];
      bb.x = W1[(k0 + kb) * 256 + tn * 16 + mn];
      bb.y = W1[(k0 + kb + 1) * 256 + tn * 16 + mn];
      acc = wmma4(a, bb, acc);
    }
#pragma unroll
    for (int r = 0; r < 8; ++r) {
      int row = r + 8 * half;
      int col = tn * 16 + mn;
      h1[row][col] = acc[r] + b1[col];
    }
  }
  __syncthreads();

  // GroupNorm(32 groups of 8) + ReLU, in place
  for (int i = tid; i < 16 * 32; i += 256) {
    int row = i >> 5, g = i & 31, base = g * 8;
    float mu = 0.f;
#pragma unroll
    for (int c = 0; c < 8; ++c) mu += h1[row][base + c];
    mu *= (1.f / 8.f);
    float var = 0.f;
#pragma unroll
    for (int c = 0; c < 8; ++c) {
      float d = h1[row][base + c] - mu;
      var += d * d;
    }
    var *= (1.f / 8.f);
    float inv = rsqrtf(var + EPSGN);
#pragma unroll
    for (int c = 0; c < 8; ++c) {
      int cc = base + c;
      float xn = (h1[row][cc] - mu) * inv * g1[cc] + beta1[cc];
      h1[row][cc] = fmaxf(xn, 0.f);
    }
  }
  __syncthreads();

  // GEMM2: (16x256) @ W2(256x128), 8 N-tiles, 1 per wave
  {
    int tn = wave;
    v8f acc = {0.f, 0.f, 0.f, 0.f, 0.f, 0.f, 0.f, 0.f};
    for (int k0 = 0; k0 < 256; k0 += 4) {
      v2f a, bb;
      a.x = h1[mn][k0 + kb];
      a.y = h1[mn][k0 + kb + 1];
      bb.x = W2[(k0 + kb) * 128 + tn * 16 + mn];
      bb.y = W2[(k0 + kb + 1) * 128 + tn * 16 + mn];
      acc = wmma4(a, bb, acc);
    }
#pragma unroll
    for (int r = 0; r < 8; ++r) {
      int row = r + 8 * half;
      int col = tn * 16 + mn;
      h2[row][col] = acc[r] + b2[col];
    }
  }
  __syncthreads();

  // GroupNorm(32 groups of 4) + ReLU, in place
  for (int i = tid; i < 16 * 32; i += 256) {
    int row = i >> 5, g = i & 31, base = g * 4;
    float mu = 0.f;
#pragma unroll
    for (int c = 0; c < 4; ++c) mu += h2[row][base + c];
    mu *= 0.25f;
    float var = 0.f;
#pragma unroll
    for (int c = 0; c < 4; ++c) {
      float d = h2[row][base + c] - mu;
      var += d * d;
    }
    var *= 0.25f;
    float inv = rsqrtf(var + EPSGN);
#pragma unroll
    for (int c = 0; c < 4; ++c) {
      int cc = base + c;
      float xn = (h2[row][cc] - mu) * inv * g2[cc] + beta2[cc];
      h2[row][cc] = fmaxf(xn, 0.f);
    }
  }
  __syncthreads();

  // Final (16x128) @ W3(128x2) + b3
  if (tid < 32) {
    int row = tid >> 1, col = tid & 1;
    float s = b3[col];
    for (int k = 0; k < 128; ++k) s += h2[row][k] * W3[k * 2 + col];
    out[(blk * 16 + row) * 2 + col] = s;
  }
}

extern "C" void kernel_launch(void* const* d_in, const int* in_sizes, int n_in,
                              void* d_out, int out_size, void* d_ws,
                              size_t ws_size, hipStream_t stream) {
  (void)n_in; (void)out_size; (void)ws_size;
  const float* points    = (const float*)d_in[0];
  const float* trans     = (const float*)d_in[1];
  const float* ref_feats = (const float*)d_in[2];
  const float* src_feats = (const float*)d_in[3];
  const float* W1 = (const float*)d_in[4];
  const float* b1 = (const float*)d_in[5];
  const float* g1 = (const float*)d_in[6];
  const float* be1 = (const float*)d_in[7];
  const float* W2 = (const float*)d_in[8];
  const float* b2 = (const float*)d_in[9];
  const float* g2 = (const float*)d_in[10];
  const float* be2 = (const float*)d_in[11];
  const float* W3 = (const float*)d_in[12];
  const float* b3 = (const float*)d_in[13];

  int B = in_sizes[1] / 16;  // trans is (B,4,4)

  // workspace layout (floats)
  float* wsf = (float*)d_ws;
  float* w_hdr  = wsf;                 // 4
  float* w_x2   = wsf + 4;             // 128
  float* w_refn = wsf + 132;           // 512
  float* w_sim  = wsf + 644;           // 16384
  float* w_sms  = wsf + 17028;         // 256
  int*   wsi    = (int*)(wsf + 17284);
  int* w_r2si  = wsi;                  // 128
  int* w_s2ri  = wsi + 128;            // 128
  int* w_permf = wsi + 256;            // 256
  float* w_joint = wsf + 17796;        // B*512

  k_precompute<<<1, 256, 0, stream>>>(points, ref_feats, src_feats, w_hdr,
                                      w_x2, w_refn, w_sim, w_sms, w_r2si,
                                      w_s2ri, w_permf);
  k_batch<<<B, 256, 0, stream>>>(points, trans, w_hdr, w_x2, w_refn, w_sim,
                                 w_sms, w_r2si, w_s2ri, w_permf, w_joint);
  k_mlp<<<B / 16, 256, 0, stream>>>(w_joint, W1, b1, g1, be1, W2, b2, g2, be2,
                                    W3, b3, (float*)d_out);
}